// HybridBlock_10634339025371
// MI455X (gfx1250) — compile-verified
//
#include <hip/hip_runtime.h>
#include <hip/hip_bf16.h>
#include <math.h>

// ---- problem constants ----
#define BB   2
#define LL   2048          // T*W = 64*32
#define DD   512
#define HH   8
#define EE   64
#define DIN  1024
#define NST  64
#define RR   32
#define NROW (BB*LL)       // 4096

typedef float v2f __attribute__((ext_vector_type(2)));
typedef float v8f __attribute__((ext_vector_type(8)));

__device__ __forceinline__ v8f wmma4(v2f a, v2f b, v8f c) {
  // D = A(16x4,f32) * B(4x16,f32) + C(16x16,f32)
  return __builtin_amdgcn_wmma_f32_16x16x4_f32(false, a, false, b, (short)0, c, false, false);
}

// =====================================================================
// Generic GEMM: C[M x NTcols] = A[M x K] * W[N x K]^T (+bias)(+softplus)
// wave: 16 rows x NT*16 cols. block = 8 waves = 128 rows.
// grid = (num 64-col blocks, M/128). Branch-free inner loop, K unrolled
// by 16 so loads are batched ahead of the WMMA group.
// =====================================================================
template <int NT>
__global__ __launch_bounds__(256)
void gemm_wt_kernel(const float* __restrict__ A, int lda,
                    const float* __restrict__ W, int ldw,
                    const float* __restrict__ bias,
                    float* __restrict__ C, int ldc,
                    int K, int act, int colbase)
{
  const int lane = threadIdx.x & 31;
  const int wave = threadIdx.x >> 5;
  const int m16  = lane & 15;
  const int lh   = lane >> 4;           // which 16-lane half
  const int row0 = blockIdx.y * 128 + wave * 16;
  const int col0 = colbase + blockIdx.x * (NT * 16);

  const float* Arow = A + (size_t)(row0 + m16) * (size_t)lda + lh * 2;

  const float* Wp[NT];
  v8f acc[NT];
  #pragma unroll
  for (int t = 0; t < NT; ++t) {
    Wp[t]  = W + (size_t)(col0 + t * 16 + m16) * (size_t)ldw + lh * 2;
    acc[t] = (v8f){0.f,0.f,0.f,0.f,0.f,0.f,0.f,0.f};
  }

  for (int k = 0; k < K; k += 16) {
    v2f a[4];
    v2f bf[4][NT];
    #pragma unroll
    for (int s = 0; s < 4; ++s) a[s] = *(const v2f*)(Arow + k + s * 4);
    #pragma unroll
    for (int s = 0; s < 4; ++s)
      #pragma unroll
      for (int t = 0; t < NT; ++t)
        bf[s][t] = *(const v2f*)(Wp[t] + k + s * 4);
    #pragma unroll
    for (int s = 0; s < 4; ++s)
      #pragma unroll
      for (int t = 0; t < NT; ++t)
        acc[t] = wmma4(a[s], bf[s][t], acc[t]);
  }

  #pragma unroll
  for (int t = 0; t < NT; ++t) {
    const int col = col0 + t * 16 + m16;
    const float bv = bias ? bias[col] : 0.0f;
    #pragma unroll
    for (int j = 0; j < 8; ++j) {
      const int row = row0 + j + lh * 8;
      float v = acc[t][j] + bv;
      if (act == 1) v = (v > 20.0f) ? v : log1pf(__expf(v));  // softplus
      C[(size_t)row * (size_t)ldc + col] = v;
    }
  }
}

// =====================================================================
// Flash attention, causal, e=64, fp32 WMMA. One wave = one 16-row Q tile.
// q/k/v/ctx layout: [B][L][D] with head offset h*64.
// grid = B*H*(L/16)/8 = 256 blocks of 256 threads.
// =====================================================================
__global__ __launch_bounds__(256)
void attn_kernel(const float* __restrict__ q, const float* __restrict__ k,
                 const float* __restrict__ v, float* __restrict__ ctx)
{
  __shared__ __align__(16) float pl[8 * 16 * 18];   // per-wave 16x16 P tile (pad 18)
  const int lane = threadIdx.x & 31;
  const int wave = threadIdx.x >> 5;
  const int m16  = lane & 15;
  const int lh   = lane >> 4;

  const int gw = blockIdx.x * 8 + wave;
  const int qt = gw & 127;
  const int bh = gw >> 7;
  const int b  = bh >> 3;
  const int h  = bh & 7;
  const int qbase = qt * 16;
  const size_t base = (size_t)b * LL * DD + (size_t)h * EE;

  // preload Q fragments (16 k-steps over E=64)
  v2f qa[16];
  {
    const float* qp = q + base + (size_t)(qbase + m16) * DD + lh * 2;
    #pragma unroll
    for (int kk = 0; kk < 16; ++kk) qa[kk] = *(const v2f*)(qp + kk * 4);
  }

  v8f o[4];
  #pragma unroll
  for (int t = 0; t < 4; ++t) o[t] = (v8f){0.f,0.f,0.f,0.f,0.f,0.f,0.f,0.f};
  float mrow[8], lrow[8];
  #pragma unroll
  for (int j = 0; j < 8; ++j) { mrow[j] = -3.0e38f; lrow[j] = 0.0f; }

  float* myP = pl + wave * 16 * 18;

  for (int kv = 0; kv <= qbase; kv += 16) {
    // S = Q Kt (16x16): two independent WMMA chains for ILP
    v8f s0 = (v8f){0.f,0.f,0.f,0.f,0.f,0.f,0.f,0.f};
    v8f s1 = (v8f){0.f,0.f,0.f,0.f,0.f,0.f,0.f,0.f};
    {
      const float* kp = k + base + (size_t)(kv + m16) * DD + lh * 2;
      #pragma unroll
      for (int kk = 0; kk < 8; ++kk) {
        v2f b0 = *(const v2f*)(kp + (2 * kk) * 4);
        v2f b1 = *(const v2f*)(kp + (2 * kk + 1) * 4);
        s0 = wmma4(qa[2 * kk],     b0, s0);
        s1 = wmma4(qa[2 * kk + 1], b1, s1);
      }
    }
    v8f s = s0 + s1;
    const bool diag = (kv == qbase);
    #pragma unroll
    for (int j = 0; j < 8; ++j) {
      float sv = s[j] * 0.125f;                       // 1/sqrt(64)
      if (diag && (kv + m16) > (qbase + j + lh * 8)) sv = -3.0e38f;
      s[j] = sv;
    }
    // online softmax; row j+lh*8 lives in one 16-lane half
    float alpha[8];
    #pragma unroll
    for (int j = 0; j < 8; ++j) {
      float rm = s[j];
      rm = fmaxf(rm, __shfl_xor(rm, 1, 32));
      rm = fmaxf(rm, __shfl_xor(rm, 2, 32));
      rm = fmaxf(rm, __shfl_xor(rm, 4, 32));
      rm = fmaxf(rm, __shfl_xor(rm, 8, 32));
      const float mn = fmaxf(mrow[j], rm);
      alpha[j] = __expf(mrow[j] - mn);
      mrow[j] = mn;
      float p = __expf(s[j] - mn);
      s[j] = p;
      float rs = p;
      rs += __shfl_xor(rs, 1, 32);
      rs += __shfl_xor(rs, 2, 32);
      rs += __shfl_xor(rs, 4, 32);
      rs += __shfl_xor(rs, 8, 32);
      lrow[j] = lrow[j] * alpha[j] + rs;
    }
    #pragma unroll
    for (int t = 0; t < 4; ++t)
      #pragma unroll
      for (int j = 0; j < 8; ++j) o[t][j] *= alpha[j];

    // P -> LDS (C-layout to A-layout transpose via LDS)
    #pragma unroll
    for (int j = 0; j < 8; ++j) myP[(j + lh * 8) * 18 + m16] = s[j];
    asm volatile("s_wait_dscnt 0" ::: "memory");       // DS in-order per wave

    // O += P V   (K-dim = 16, 4 WMMA k-steps x 4 e-tiles, 4 indep chains)
    const float* vp = v + base + (size_t)kv * DD;
    #pragma unroll
    for (int kk = 0; kk < 4; ++kk) {
      v2f af = *(const v2f*)(myP + m16 * 18 + kk * 4 + lh * 2);
      #pragma unroll
      for (int t = 0; t < 4; ++t) {
        const float* b0 = vp + (size_t)(kk * 4 + lh * 2) * DD + t * 16 + m16;
        v2f bfr; bfr.x = b0[0]; bfr.y = b0[DD];
        o[t] = wmma4(af, bfr, o[t]);
      }
    }
    asm volatile("s_wait_dscnt 0" ::: "memory");
  }

  float inv[8];
  #pragma unroll
  for (int j = 0; j < 8; ++j) inv[j] = 1.0f / lrow[j];
  float* cp = ctx + base + (size_t)qbase * DD;
  #pragma unroll
  for (int t = 0; t < 4; ++t)
    #pragma unroll
    for (int j = 0; j < 8; ++j)
      cp[(size_t)(j + lh * 8) * DD + t * 16 + m16] = o[t][j] * inv[j];
}

// =====================================================================
// depthwise causal conv1d (K=4) over L + SiLU. x_in is cols [0,DIN) of xz.
// =====================================================================
__global__ __launch_bounds__(256)
void conv_silu_kernel(const float* __restrict__ xz, const float* __restrict__ cw,
                      const float* __restrict__ cb, float* __restrict__ u)
{
  const size_t idx = (size_t)blockIdx.x * 256 + threadIdx.x;   // < NROW*DIN
  const int c   = (int)(idx & (DIN - 1));
  const int row = (int)(idx >> 10);
  const int b   = row >> 11;
  const int l   = row & (LL - 1);
  float acc = cb[c];
  #pragma unroll
  for (int kk = 0; kk < 4; ++kk) {
    const int ls = l - 3 + kk;
    if (ls >= 0)
      acc += xz[((size_t)(b * LL + ls)) * (2 * DIN) + c] * cw[c * 4 + kk];
  }
  u[idx] = acc / (1.0f + __expf(-acc));   // silu
}

// =====================================================================
// selective scan: one wave per (b,d) channel, 2 states/lane.
// B/C tiles staged in LDS (32 time steps) shared by 8 channels/block.
// grid = B*DIN/8 = 256 blocks.
// =====================================================================
__global__ __launch_bounds__(256)
void scan_kernel(const float* __restrict__ dbc, const float* __restrict__ dt,
                 const float* __restrict__ u, const float* __restrict__ A_log,
                 float* __restrict__ y)
{
  __shared__ float sB[32][NST];
  __shared__ float sC[32][NST];
  const int lane = threadIdx.x & 31;
  const int wave = threadIdx.x >> 5;
  const int b  = blockIdx.x >> 7;            // 128 groups per batch
  const int d  = (blockIdx.x & 127) * 8 + wave;
  const int n0 = lane * 2, n1 = lane * 2 + 1;

  const float a0 = -__expf(A_log[d * NST + n0]);
  const float a1 = -__expf(A_log[d * NST + n1]);
  float h0 = 0.0f, h1 = 0.0f;

  for (int tc = 0; tc < LL / 32; ++tc) {
    __syncthreads();
    for (int i = threadIdx.x; i < 32 * NST; i += 256) {
      const int tl = i >> 6, n = i & 63;
      const size_t rb = ((size_t)(b * LL + tc * 32 + tl)) * (RR + 2 * NST);
      sB[tl][n] = dbc[rb + RR + n];
      sC[tl][n] = dbc[rb + RR + NST + n];
    }
    __syncthreads();
    for (int tl = 0; tl < 32; ++tl) {
      const size_t t = (size_t)(b * LL + tc * 32 + tl);
      const float dtv = dt[t * DIN + d];
      const float uv  = u[t * DIN + d];
      const float du  = dtv * uv;
      h0 = __expf(dtv * a0) * h0 + du * sB[tl][n0];
      h1 = __expf(dtv * a1) * h1 + du * sB[tl][n1];
      float yp = h0 * sC[tl][n0] + h1 * sC[tl][n1];
      yp += __shfl_xor(yp, 16, 32);
      yp += __shfl_xor(yp,  8, 32);
      yp += __shfl_xor(yp,  4, 32);
      yp += __shfl_xor(yp,  2, 32);
      yp += __shfl_xor(yp,  1, 32);
      if (lane == 0) y[t * DIN + d] = yp;
    }
  }
}

// y = (y + u*D) * silu(z);  z = cols [DIN,2DIN) of xz
__global__ __launch_bounds__(256)
void ygate_kernel(float* __restrict__ y, const float* __restrict__ u,
                  const float* __restrict__ Dp, const float* __restrict__ xz)
{
  const size_t idx = (size_t)blockIdx.x * 256 + threadIdx.x;
  const int d   = (int)(idx & (DIN - 1));
  const size_t row = idx >> 10;
  const float z = xz[row * (2 * DIN) + DIN + d];
  const float sil = z / (1.0f + __expf(-z));
  y[idx] = (y[idx] + u[idx] * Dp[d]) * sil;
}

// residual sum + layernorm; one block per row of 512
__global__ __launch_bounds__(256)
void ln_kernel(const float* __restrict__ x, const float* __restrict__ xa,
               const float* __restrict__ xm, const float* __restrict__ w,
               const float* __restrict__ bta, float* __restrict__ out)
{
  __shared__ float red[256];
  const size_t rb = (size_t)blockIdx.x * DD;
  const int t = threadIdx.x;
  const float h0 = x[rb + t]       + xa[rb + t]       + xm[rb + t];
  const float h1 = x[rb + t + 256] + xa[rb + t + 256] + xm[rb + t + 256];
  red[t] = h0 + h1;
  __syncthreads();
  for (int off = 128; off > 0; off >>= 1) {
    if (t < off) red[t] += red[t + off];
    __syncthreads();
  }
  const float mu = red[0] * (1.0f / DD);
  __syncthreads();
  const float d0 = h0 - mu, d1 = h1 - mu;
  red[t] = d0 * d0 + d1 * d1;
  __syncthreads();
  for (int off = 128; off > 0; off >>= 1) {
    if (t < off) red[t] += red[t + off];
    __syncthreads();
  }
  const float r = rsqrtf(red[0] * (1.0f / DD) + 1e-5f);
  out[rb + t]       = d0 * r * w[t]       + bta[t];
  out[rb + t + 256] = d1 * r * w[t + 256] + bta[t + 256];
}

// =====================================================================
extern "C" void kernel_launch(void* const* d_in, const int* in_sizes, int n_in,
                              void* d_out, int out_size, void* d_ws, size_t ws_size,
                              hipStream_t stream)
{
  const float* x         = (const float*)d_in[0];
  const float* wq        = (const float*)d_in[1];
  const float* bq        = (const float*)d_in[2];
  const float* wk        = (const float*)d_in[3];
  const float* bk        = (const float*)d_in[4];
  const float* wv        = (const float*)d_in[5];
  const float* bv        = (const float*)d_in[6];
  const float* wo        = (const float*)d_in[7];
  const float* bo        = (const float*)d_in[8];
  const float* in_proj_w = (const float*)d_in[9];
  const float* conv_w    = (const float*)d_in[10];
  const float* conv_b    = (const float*)d_in[11];
  const float* x_proj_w  = (const float*)d_in[12];
  const float* dt_proj_w = (const float*)d_in[13];
  const float* dt_proj_b = (const float*)d_in[14];
  const float* A_log     = (const float*)d_in[15];
  const float* D_param   = (const float*)d_in[16];
  const float* out_proj_w= (const float*)d_in[17];
  const float* ln_w      = (const float*)d_in[18];
  const float* ln_b      = (const float*)d_in[19];
  float* out = (float*)d_out;

  float* ws = (float*)d_ws;
  const size_t SZ = (size_t)NROW * DD;            // 2,097,152
  float* q      = ws;
  float* k      = ws + 1 * SZ;
  float* v      = ws + 2 * SZ;
  float* ctx    = ws + 3 * SZ;
  float* xattn  = ws + 4 * SZ;
  float* xz     = ws + 5 * SZ;                    // NROW x 2048
  float* u      = ws + 9 * SZ;                    // NROW x 1024
  float* dt     = ws + 11 * SZ;                   // NROW x 1024
  float* y      = ws + 13 * SZ;                   // NROW x 1024
  float* dbc    = ws + 15 * SZ;                   // NROW x 160
  float* xmamba = ws + 15 * SZ + (size_t)NROW * (RR + 2 * NST);

  dim3 blk(256);
  // QKV projections
  gemm_wt_kernel<4><<<dim3(8, 32),  blk, 0, stream>>>(x, DD, wq, DD, bq, q, DD, DD, 0, 0);
  gemm_wt_kernel<4><<<dim3(8, 32),  blk, 0, stream>>>(x, DD, wk, DD, bk, k, DD, DD, 0, 0);
  gemm_wt_kernel<4><<<dim3(8, 32),  blk, 0, stream>>>(x, DD, wv, DD, bv, v, DD, DD, 0, 0);
  // attention
  attn_kernel<<<dim3(256), blk, 0, stream>>>(q, k, v, ctx);
  gemm_wt_kernel<4><<<dim3(8, 32),  blk, 0, stream>>>(ctx, DD, wo, DD, bo, xattn, DD, DD, 0, 0);
  // mamba branch
  gemm_wt_kernel<4><<<dim3(32, 32), blk, 0, stream>>>(x, DD, in_proj_w, DD, nullptr, xz, 2 * DIN, DD, 0, 0);
  conv_silu_kernel<<<dim3((NROW * DIN) / 256), blk, 0, stream>>>(xz, conv_w, conv_b, u);
  // x_proj: N = 160 = 2 full 64-col blocks + one 32-col remainder
  gemm_wt_kernel<4><<<dim3(2, 32),  blk, 0, stream>>>(u, DIN, x_proj_w, DIN, nullptr, dbc, RR + 2 * NST, DIN, 0, 0);
  gemm_wt_kernel<2><<<dim3(1, 32),  blk, 0, stream>>>(u, DIN, x_proj_w, DIN, nullptr, dbc, RR + 2 * NST, DIN, 0, 128);
  // dt_proj + softplus
  gemm_wt_kernel<4><<<dim3(16, 32), blk, 0, stream>>>(dbc, RR + 2 * NST, dt_proj_w, RR, dt_proj_b, dt, DIN, RR, 1, 0);
  scan_kernel<<<dim3(256), blk, 0, stream>>>(dbc, dt, u, A_log, y);
  ygate_kernel<<<dim3((NROW * DIN) / 256), blk, 0, stream>>>(y, u, D_param, xz);
  gemm_wt_kernel<4><<<dim3(8, 32),  blk, 0, stream>>>(y, DIN, out_proj_w, DIN, nullptr, xmamba, DD, DIN, 0, 0);
  // residual + layernorm
  ln_kernel<<<dim3(NROW), blk, 0, stream>>>(x, xattn, xmamba, ln_w, ln_b, out);
}